// MaskRCNN_39659728011751
// MI455X (gfx1250) — compile-verified
//
#include <hip/hip_runtime.h>
#include <hip/hip_bf16.h>

#define GT_CAP   64
#define MASK_SZ  28
#define MASK_PIX (MASK_SZ * MASK_SZ)
#define IMG      800

typedef float v2f __attribute__((ext_vector_type(2)));
typedef float v8f __attribute__((ext_vector_type(8)));

// ---------------------------------------------------------------------------
// Exact f32 wave32 all-reduce using V_WMMA_F32_16X16X4_F32.
// A 16x4 layout: vgpr0 = K0 (lanes0-15) / K2 (lanes16-31); vgpr1 = K1/K3.
// With a = {p, 0} and B = ones: D[m][n] = p[m] + p[m+16] for every n.
// D layout: vgpr j holds rows M=j (lanes0-15) and M=8+j (lanes16-31), so
// sum(d0..d7) gives half-sum per 16-lane group; one xor-16 shuffle finishes.
// ---------------------------------------------------------------------------
__device__ __forceinline__ float wave_sum_wmma(float v) {
  v2f a; a[0] = v;   a[1] = 0.0f;
  v2f b; b[0] = 1.0f; b[1] = 1.0f;
  v8f c = {};
  v8f d = __builtin_amdgcn_wmma_f32_16x16x4_f32(false, a, false, b,
                                                (short)0, c, false, false);
  float s = d[0] + d[1] + d[2] + d[3] + d[4] + d[5] + d[6] + d[7];
  s += __shfl_xor(s, 16, 32);
  return s;  // total in every lane
}

// Deterministic block reduction (valid result on thread 0 only).
__device__ __forceinline__ float block_sum(float v, float* red) {
  float w = wave_sum_wmma(v);
  const int lane = threadIdx.x & 31;
  const int wid  = threadIdx.x >> 5;
  __syncthreads();
  if (lane == 0) red[wid] = w;
  __syncthreads();
  float t = 0.0f;
  if (threadIdx.x == 0) {
    const int nw = blockDim.x >> 5;
    for (int i = 0; i < nw; ++i) t += red[i];
  }
  __syncthreads();
  return t;
}

__device__ __forceinline__ float iou_pair(float4 a, float4 g, float aa, float ga) {
  float lx = fmaxf(a.x, g.x), ly = fmaxf(a.y, g.y);
  float rx = fminf(a.z, g.z), ry = fminf(a.w, g.w);
  float w = fmaxf(rx - lx, 0.0f), h = fmaxf(ry - ly, 0.0f);
  float inter = w * h;
  return inter / (aa + ga - inter);
}

__device__ __forceinline__ float smooth_l1(float x) {
  float ax = fabsf(x);
  return (ax < 1.0f) ? 0.5f * ax * ax : ax - 0.5f;
}

__device__ __forceinline__ float bce_logits(float l, float t) {
  return fmaxf(l, 0.0f) - l * t + log1pf(expf(-fabsf(l)));
}

// ---------------------------------------------------------------------------
// Pass A: per-GT max IoU over all anchors (for force_pos tie detection).
// Wave-level max reduce, then LDS atomic, then one global atomic per gt/block.
// ---------------------------------------------------------------------------
__global__ __launch_bounds__(256)
void k_passA(const float4* __restrict__ anchors, const float4* __restrict__ gtb,
             unsigned* __restrict__ gt_max, int N, int M) {
  __shared__ unsigned smax[GT_CAP];
  __shared__ float4 sg[GT_CAP];
  __shared__ float  sga[GT_CAP];
  const int tid = threadIdx.x;
  if (tid < M) {
    smax[tid] = 0u;
    float4 g = gtb[tid];
    sg[tid]  = g;
    sga[tid] = (g.z - g.x) * (g.w - g.y);
  }
  __syncthreads();

  const int stride = gridDim.x * blockDim.x;
  for (int i = blockIdx.x * blockDim.x + tid; i - tid < N; i += stride) {
    const bool ok = i < N;                    // loop bound is block-uniform
    float4 a = ok ? anchors[i] : make_float4(0.f, 0.f, 0.f, 0.f);
    float aa = (a.z - a.x) * (a.w - a.y);
    for (int j = 0; j < M; ++j) {
      float v = ok ? iou_pair(a, sg[j], aa, sga[j]) : 0.0f;
      for (int o = 16; o; o >>= 1) v = fmaxf(v, __shfl_xor(v, o, 32));
      if ((tid & 31) == 0) atomicMax(&smax[j], __float_as_uint(v));
    }
  }
  __syncthreads();
  if (tid < M) atomicMax(&gt_max[tid], smax[tid]);
}

// ---------------------------------------------------------------------------
// Pass B: recompute per-anchor IoUs (bit-identical), build labels, fuse
// RPN classification CE + smooth-L1 regression. Per-block partials.
// partials[blk*4 + {ce_sum, valid_cnt, reg_sum, pos_cnt}]
// ---------------------------------------------------------------------------
__global__ __launch_bounds__(256)
void k_passB(const float4* __restrict__ anchors, const float4* __restrict__ gtb,
             const unsigned* __restrict__ gt_max,
             const float2* __restrict__ cls, const float4* __restrict__ bbox,
             float* __restrict__ partials, int N, int M) {
  __shared__ float4 sg[GT_CAP];
  __shared__ float  sga[GT_CAP];
  __shared__ float  sgm[GT_CAP];
  __shared__ float  red[8];
  const int tid = threadIdx.x;
  if (tid < M) {
    float4 g = gtb[tid];
    sg[tid]  = g;
    sga[tid] = (g.z - g.x) * (g.w - g.y);
    sgm[tid] = __uint_as_float(gt_max[tid]);
  }
  __syncthreads();

  float ce_sum = 0.f, valid_cnt = 0.f, reg_sum = 0.f, pos_cnt = 0.f;
  const int stride = gridDim.x * blockDim.x;
  for (int i = blockIdx.x * blockDim.x + tid; i < N; i += stride) {
    float4 a = anchors[i];
    float aa = (a.z - a.x) * (a.w - a.y);
    float best = -1.0f; int bj = 0; bool force = false;
    for (int j = 0; j < M; ++j) {
      float v = iou_pair(a, sg[j], aa, sga[j]);
      if (v > best) { best = v; bj = j; }
      force = force || (v == sgm[j]);
    }
    int lab = -1;
    if (best >= 0.7f) lab = 1;
    if (best <= 0.3f) lab = 0;
    if (force)        lab = 1;
    if (lab >= 0) {
      float2 l = cls[i];
      float m = fmaxf(l.x, l.y);
      float lse = m + logf(expf(l.x - m) + expf(l.y - m));
      ce_sum += lse - ((lab == 1) ? l.y : l.x);
      valid_cnt += 1.0f;
    }
    if (lab == 1) {
      float4 g = sg[bj];
      float rcx = (a.x + a.z) * 0.5f, rcy = (a.y + a.w) * 0.5f;
      float rw = a.z - a.x, rh = a.w - a.y;
      float gcx = (g.x + g.z) * 0.5f, gcy = (g.y + g.w) * 0.5f;
      float gw = g.z - g.x, gh = g.w - g.y;
      float t0 = (gcx - rcx) / rw, t1 = (gcy - rcy) / rh;
      float t2 = logf(gw / rw),    t3 = logf(gh / rh);
      float4 p = bbox[i];
      reg_sum += smooth_l1(p.x - t0) + smooth_l1(p.y - t1)
               + smooth_l1(p.z - t2) + smooth_l1(p.w - t3);
      pos_cnt += 1.0f;
    }
  }
  float s0 = block_sum(ce_sum,    red);
  float s1 = block_sum(valid_cnt, red);
  float s2 = block_sum(reg_sum,   red);
  float s3 = block_sum(pos_cnt,   red);
  if (tid == 0) {
    partials[blockIdx.x * 4 + 0] = s0;
    partials[blockIdx.x * 4 + 1] = s1;
    partials[blockIdx.x * 4 + 2] = s2;
    partials[blockIdx.x * 4 + 3] = s3;
  }
}

// ---------------------------------------------------------------------------
// Stage 2: proposal matching, fg/bg sampling (needs ordered cumsum -> LDS
// scan), detection CE + regression; emits per-roi gidx/keep/mlabel for masks.
// out3 = {det_cls_loss, det_reg_loss, keep_sum}
// ---------------------------------------------------------------------------
__global__ __launch_bounds__(1024)
void k_stage2(const float4* __restrict__ props, const float4* __restrict__ gtb,
              const int* __restrict__ gtl,
              const float* __restrict__ cls, const float* __restrict__ deltas,
              float* __restrict__ out3, int* __restrict__ o_gidx,
              int* __restrict__ o_keep, int* __restrict__ o_mlab,
              int K, int M, int C) {
  __shared__ float4 sg[GT_CAP];
  __shared__ float  sga[GT_CAP];
  __shared__ int    sgl[GT_CAP];
  __shared__ int    scF[1024];
  __shared__ int    scB[1024];
  __shared__ float  red[32];
  const int tid = threadIdx.x;
  if (tid < M) {
    float4 g = gtb[tid];
    sg[tid]  = g;
    sga[tid] = (g.z - g.x) * (g.w - g.y);
    sgl[tid] = gtl[tid];
  }
  __syncthreads();

  const bool in = tid < K;
  float miou = 0.0f; int gi = 0; int plab = -1;
  float4 pr = make_float4(0.f, 0.f, 0.f, 0.f);
  if (in) {
    pr = props[tid];
    float pa = (pr.z - pr.x) * (pr.w - pr.y);
    float best = -1.0f; int bj = 0;
    for (int j = 0; j < M; ++j) {
      float v = iou_pair(pr, sg[j], pa, sga[j]);
      if (v > best) { best = v; bj = j; }
    }
    miou = best; gi = bj;
    plab = sgl[bj];
    if (miou < 0.5f) plab = 0;
    if (miou < 0.1f) plab = -1;
  }
  const int fg = (in && plab > 0)  ? 1 : 0;
  const int bg = (in && plab == 0) ? 1 : 0;
  scF[tid] = fg; scB[tid] = bg;
  __syncthreads();
  for (int off = 1; off < 1024; off <<= 1) {     // inclusive Hillis-Steele
    int a = (tid >= off) ? scF[tid - off] : 0;
    int b = (tid >= off) ? scB[tid - off] : 0;
    __syncthreads();
    scF[tid] += a; scB[tid] += b;
    __syncthreads();
  }
  const int nfg = min(32, scF[1023]);
  const int nbg = 128 - nfg;
  const int selfg = fg && (scF[tid] <= nfg);
  const int selbg = bg && (scB[tid] <= nbg);
  const int sel = selfg || selbg;

  float ce = 0.0f;
  if (in && sel) {
    const float* row = cls + tid * C;
    float m = row[0];
    for (int c = 1; c < C; ++c) m = fmaxf(m, row[c]);
    float s = 0.0f;
    for (int c = 0; c < C; ++c) s += expf(row[c] - m);
    ce = (m + logf(s)) - row[max(plab, 0)];
  }
  float reg = 0.0f;
  if (in && selfg) {
    float4 g = sg[gi];
    float rcx = (pr.x + pr.z) * 0.5f, rcy = (pr.y + pr.w) * 0.5f;
    float rw = pr.z - pr.x, rh = pr.w - pr.y;
    float gcx = (g.x + g.z) * 0.5f, gcy = (g.y + g.w) * 0.5f;
    float gw = g.z - g.x, gh = g.w - g.y;
    float t0 = (gcx - rcx) / rw, t1 = (gcy - rcy) / rh;
    float t2 = logf(gw / rw),    t3 = logf(gh / rh);
    const float* dr = deltas + tid * C * 4 + max(plab, 0) * 4;
    reg = smooth_l1(dr[0] - t0) + smooth_l1(dr[1] - t1)
        + smooth_l1(dr[2] - t2) + smooth_l1(dr[3] - t3);
  }
  const int keep = (in && miou >= 0.5f) ? 1 : 0;
  if (in) { o_gidx[tid] = gi; o_keep[tid] = keep; o_mlab[tid] = sgl[gi]; }

  float S_ce   = block_sum(ce,            red);
  float S_sel  = block_sum((float)sel,    red);
  float S_reg  = block_sum(reg,           red);
  float S_fg   = block_sum((float)selfg,  red);
  float S_keep = block_sum((float)keep,   red);
  if (tid == 0) {
    out3[0] = S_ce  / fmaxf(S_sel, 1.0f);
    out3[1] = S_reg / fmaxf(S_fg * 4.0f, 1.0f);
    out3[2] = S_keep;
  }
}

// ---------------------------------------------------------------------------
// Mask branch: ROI-align bilinear gather (shared coords for vis/amodal) +
// BCE-with-logits; skip keep==0 (contribution exactly 0). Per-block partials.
// ---------------------------------------------------------------------------
__global__ __launch_bounds__(256)
void k_mask(const float* __restrict__ vism, const float* __restrict__ amom,
            const float* __restrict__ vlog, const float* __restrict__ alog,
            const float4* __restrict__ props,
            const int* __restrict__ gidx, const int* __restrict__ keep,
            const int* __restrict__ mlab,
            float* __restrict__ partials, int K, int C) {
  __shared__ float red[8];
  const int e = blockIdx.x * 256 + threadIdx.x;
  float contrib = 0.0f;
  if (e < K * MASK_PIX) {
    const int i = e / MASK_PIX, p = e % MASK_PIX;
    if (keep[i]) {
      const int py = p / MASK_SZ, px = p % MASK_SZ;
      float4 r = props[i];
      float bw = (r.z - r.x) * (1.0f / MASK_SZ);
      float bh = (r.w - r.y) * (1.0f / MASK_SZ);
      float X = (r.x - 0.5f) + ((float)px + 0.5f) * bw;
      float Y = (r.y - 0.5f) + ((float)py + 0.5f) * bh;
      bool valid = (Y > -1.0f) && (Y < (float)IMG) && (X > -1.0f) && (X < (float)IMG);
      float Yc = fminf(fmaxf(Y, 0.0f), (float)(IMG - 1));
      float Xc = fminf(fmaxf(X, 0.0f), (float)(IMG - 1));
      int y0 = (int)floorf(Yc), x0 = (int)floorf(Xc);
      int y1 = min(y0 + 1, IMG - 1), x1 = min(x0 + 1, IMG - 1);
      float ly = Yc - (float)y0, lx = Xc - (float)x0;
      float w00 = (1.f - ly) * (1.f - lx), w01 = (1.f - ly) * lx;
      float w10 = ly * (1.f - lx),         w11 = ly * lx;
      int base = gidx[i] * (IMG * IMG);
      int i00 = base + y0 * IMG + x0, i01 = base + y0 * IMG + x1;
      int i10 = base + y1 * IMG + x0, i11 = base + y1 * IMG + x1;
      float tv = valid ? (vism[i00]*w00 + vism[i01]*w01 + vism[i10]*w10 + vism[i11]*w11) : 0.0f;
      float ta = valid ? (amom[i00]*w00 + amom[i01]*w01 + amom[i10]*w10 + amom[i11]*w11) : 0.0f;
      int lidx = (i * C + mlab[i]) * MASK_PIX + p;
      contrib = bce_logits(vlog[lidx], tv) + bce_logits(alog[lidx], ta);
    }
  }
  float t = block_sum(contrib, red);
  if (threadIdx.x == 0) partials[blockIdx.x] = t;
}

// ---------------------------------------------------------------------------
// Final combine: deterministic reduction of all partials -> scalar loss.
// ---------------------------------------------------------------------------
__global__ __launch_bounds__(256)
void k_final(const float* __restrict__ passBp, int nB,
             const float* __restrict__ maskP, int nM,
             const float* __restrict__ st2, float* __restrict__ out) {
  __shared__ float red[8];
  float a0 = 0.f, a1 = 0.f, a2 = 0.f, a3 = 0.f, am = 0.f;
  for (int r = threadIdx.x; r < nB; r += blockDim.x) {
    a0 += passBp[r * 4 + 0]; a1 += passBp[r * 4 + 1];
    a2 += passBp[r * 4 + 2]; a3 += passBp[r * 4 + 3];
  }
  for (int r = threadIdx.x; r < nM; r += blockDim.x) am += maskP[r];
  float S_ce  = block_sum(a0, red);
  float S_val = block_sum(a1, red);
  float S_reg = block_sum(a2, red);
  float S_pos = block_sum(a3, red);
  float S_m   = block_sum(am, red);
  if (threadIdx.x == 0) {
    float rpn_cls  = S_ce  / fmaxf(S_val, 1.0f);
    float rpn_reg  = S_reg / fmaxf(S_pos * 4.0f, 1.0f);
    float mask     = S_m / (fmaxf(st2[2], 1.0f) * (float)MASK_PIX);
    out[0] = rpn_cls + rpn_reg + st2[0] + st2[1] + mask;
  }
}

// ---------------------------------------------------------------------------
extern "C" void kernel_launch(void* const* d_in, const int* in_sizes, int n_in,
                              void* d_out, int out_size, void* d_ws, size_t ws_size,
                              hipStream_t stream) {
  const float* rpn_cls   = (const float*)d_in[0];
  const float* rpn_bbox  = (const float*)d_in[1];
  const float* anchors   = (const float*)d_in[2];
  const float* gt_boxes  = (const float*)d_in[3];
  const int*   gt_labels = (const int*)  d_in[4];
  const float* roi_cls   = (const float*)d_in[5];
  const float* roi_delta = (const float*)d_in[6];
  const float* proposals = (const float*)d_in[7];
  const float* mvl       = (const float*)d_in[8];
  const float* mal       = (const float*)d_in[9];
  const float* vism      = (const float*)d_in[10];
  const float* amom      = (const float*)d_in[11];

  const int N = in_sizes[2] / 4;   // anchors
  const int M = in_sizes[3] / 4;   // gt boxes (<= GT_CAP)
  const int K = in_sizes[7] / 4;   // proposals
  const int C = in_sizes[5] / K;   // classes

  const int NB = 480;                              // passA/passB blocks
  const int NM = (K * MASK_PIX + 255) / 256;       // mask blocks

  unsigned char* w = (unsigned char*)d_ws;
  unsigned* gt_max = (unsigned*)w;  w += GT_CAP * sizeof(unsigned);
  float* passBp    = (float*)w;     w += (size_t)NB * 4 * sizeof(float);
  float* st2       = (float*)w;     w += 4 * sizeof(float);
  int*   o_gidx    = (int*)w;       w += (size_t)K * sizeof(int);
  int*   o_keep    = (int*)w;       w += (size_t)K * sizeof(int);
  int*   o_mlab    = (int*)w;       w += (size_t)K * sizeof(int);
  float* maskP     = (float*)w;     w += (size_t)NM * sizeof(float);

  hipMemsetAsync(gt_max, 0, GT_CAP * sizeof(unsigned), stream);

  k_passA<<<NB, 256, 0, stream>>>((const float4*)anchors, (const float4*)gt_boxes,
                                  gt_max, N, M);
  k_passB<<<NB, 256, 0, stream>>>((const float4*)anchors, (const float4*)gt_boxes,
                                  gt_max, (const float2*)rpn_cls,
                                  (const float4*)rpn_bbox, passBp, N, M);
  k_stage2<<<1, 1024, 0, stream>>>((const float4*)proposals, (const float4*)gt_boxes,
                                   gt_labels, roi_cls, roi_delta,
                                   st2, o_gidx, o_keep, o_mlab, K, M, C);
  k_mask<<<NM, 256, 0, stream>>>(vism, amom, mvl, mal, (const float4*)proposals,
                                 o_gidx, o_keep, o_mlab, maskP, K, C);
  k_final<<<1, 256, 0, stream>>>(passBp, NB, maskP, NM, st2, (float*)d_out);
}